// HyperbolicGraphConvolution_67568425500962
// MI455X (gfx1250) — compile-verified
//
#include <hip/hip_runtime.h>
#include <hip/hip_bf16.h>

// ---------------------------------------------------------------------------
// HyperbolicGraphConvolution for MI455X (gfx1250, wave32, WMMA).
// fp32 WMMA (V_WMMA_F32_16X16X4_F32) GEMMs with branch-free, software-
// pipelined inner loops:
//  - each wave owns an interleaved K-slice and drives 4 independent
//    accumulator chains (4 N-tiles) from 1 A-load + 4 B-loads per step
//  - 1-deep pipeline: fragment k+16 is prefetched before fragment k is fed
//    to the WMMA burst (loads stay 1 iteration ahead of consumption)
//  - OOB rows handled by pointer clamping; K-tail peeled out of the hot loop
//  - cross-wave partial-tile reduction through LDS
// ---------------------------------------------------------------------------

typedef float v2f __attribute__((ext_vector_type(2)));
typedef float v8f __attribute__((ext_vector_type(8)));

#define N_NODES 1546
#define DIM_IN  1546
#define DIM_OUT 64
#define N_DRUG  1373
#define M_TILES 97            // ceil(1546/16)
#define K_MAIN  1544          // 1546 rounded down to multiple of 8 (= 386*4)
#define MAXNORM (1.0f - 4e-3f)
#define MIN_NRM 1e-15f

__device__ __forceinline__ float artanh_f(float v) {
  v = fminf(v, 1.0f - 1e-7f);
  v = fmaxf(v, -1.0f + 1e-7f);
  return 0.5f * (log1pf(v) - log1pf(-v));
}

__device__ __forceinline__ v8f wmma_f32_k4(v2f a, v2f b, v8f c) {
  // D = A(16x4,f32) * B(4x16,f32) + C(16x16,f32), wave32.
  return __builtin_amdgcn_wmma_f32_16x16x4_f32(false, a, false, b, (short)0, c,
                                               false, false);
}

// ---------------------------------------------------------------------------
// Kernel 1: HypLinear.  One block = one 16-row strip.  Wave w covers the
// K-slice {w*4 + 16*i} and all 64 output columns (4 acc chains).
// ---------------------------------------------------------------------------
__global__ __launch_bounds__(128)
void hgc_hyp_linear(const float* __restrict__ x,
                    const float* __restrict__ Wd,
                    const float* __restrict__ Wm,
                    const float* __restrict__ bd,
                    const float* __restrict__ bm,
                    float* __restrict__ liner_h,
                    float* __restrict__ x_tan) {
  __shared__ float s_part[4][16][65];     // per-wave partial mx tiles
  __shared__ float s_mx[16][65];          // reduced mx tile
  __shared__ float s_braw[2][64];
  __shared__ float s_hb[2][64];           // hyp_bias = proj(expmap0(bias))
  __shared__ float s_hbg[2], s_hby2[2];
  __shared__ float s_sxp[4][16];          // per-wave ||x_row||^2 partials
  __shared__ float s_ps[16][8], s_pd[16][8];
  __shared__ float s_A1[16], s_A2[16], s_T1[16], s_T2[16];

  const int t    = threadIdx.x;
  const int lane = t & 31;
  const int wave = t >> 5;
  const int half = lane >> 4;             // K sub-offset of the fragment
  const int l16  = lane & 15;
  const int row_base = blockIdx.x * 16;

  // ---- hyperbolic bias (both classes) ----
  if (t < 64) { s_braw[0][t] = bd[t]; s_braw[1][t] = bm[t]; }
  __syncthreads();
  if (t < 2) {
    float ss = 0.f;
    for (int c = 0; c < 64; ++c) { float v = s_braw[t][c]; ss += v * v; }
    float na = sqrtf(ss);
    float un = fmaxf(na, MIN_NRM);
    float e  = tanhf(un) / un;
    float nh = e * na;
    if (nh > MAXNORM) { e *= MAXNORM / nh; nh = MAXNORM; }
    s_hbg[t]  = e;
    s_hby2[t] = nh * nh;
  }
  __syncthreads();
  { int cls = t >> 6, c = t & 63; s_hb[cls][c] = s_hbg[cls] * s_braw[cls][c]; }

  // A-row pointer, clamped (garbage rows never stored)
  const int row_u  = row_base + l16;
  const int row_a  = (row_u < N_NODES) ? row_u : (N_NODES - 1);
  const float* xrow = x + (size_t)row_a * DIM_IN;
  const int c0 = l16, c1 = 16 + l16, c2 = 32 + l16, c3 = 48 + l16;

  const bool needD = row_base < N_DRUG;
  const bool needM = (row_base + 15) >= N_DRUG;

  for (int wsel = 0; wsel < 2; ++wsel) {  // uniform per block
    if (wsel == 0 && !needD) continue;
    if (wsel == 1 && !needM) continue;
    const float* W  = (wsel == 0) ? Wd : Wm;
    const float* w0 = W + (size_t)c0 * DIM_IN;   // B[k][n] = W[n][k]
    const float* w1 = W + (size_t)c1 * DIM_IN;
    const float* w2 = W + (size_t)c2 * DIM_IN;
    const float* w3 = W + (size_t)c3 * DIM_IN;

    v8f acc0 = {0.f,0.f,0.f,0.f,0.f,0.f,0.f,0.f};
    v8f acc1 = acc0, acc2 = acc0, acc3 = acc0;
    float sumsq = 0.f;

    // ---- software-pipelined main loop over this wave's K-slice ----
    int k = wave * 4;                     // every wave runs >= 1 iteration
    v2f a  = *(const v2f*)(xrow + k + half * 2);
    v2f b0 = *(const v2f*)(w0 + k + half * 2);
    v2f b1 = *(const v2f*)(w1 + k + half * 2);
    v2f b2 = *(const v2f*)(w2 + k + half * 2);
    v2f b3 = *(const v2f*)(w3 + k + half * 2);
    for (; k + 16 < K_MAIN; k += 16) {
      const int kn = k + 16 + half * 2;   // prefetch next fragment first
      const v2f a_n  = *(const v2f*)(xrow + kn);
      const v2f b0_n = *(const v2f*)(w0 + kn);
      const v2f b1_n = *(const v2f*)(w1 + kn);
      const v2f b2_n = *(const v2f*)(w2 + kn);
      const v2f b3_n = *(const v2f*)(w3 + kn);
      sumsq = fmaf(a.x, a.x, sumsq);
      sumsq = fmaf(a.y, a.y, sumsq);
      acc0 = wmma_f32_k4(a, b0, acc0);
      acc1 = wmma_f32_k4(a, b1, acc1);
      acc2 = wmma_f32_k4(a, b2, acc2);
      acc3 = wmma_f32_k4(a, b3, acc3);
      a = a_n; b0 = b0_n; b1 = b1_n; b2 = b2_n; b3 = b3_n;
    }
    sumsq = fmaf(a.x, a.x, sumsq);        // epilogue: last main fragment
    sumsq = fmaf(a.y, a.y, sumsq);
    acc0 = wmma_f32_k4(a, b0, acc0);
    acc1 = wmma_f32_k4(a, b1, acc1);
    acc2 = wmma_f32_k4(a, b2, acc2);
    acc3 = wmma_f32_k4(a, b3, acc3);

    // K tail (1544..1545): wave 0 only, half==1 lanes masked to zero
    if (wave == 0) {
      v2f at = {0.f,0.f}, t0 = {0.f,0.f}, t1 = {0.f,0.f},
          t2 = {0.f,0.f}, t3 = {0.f,0.f};
      if (half == 0) {
        at = *(const v2f*)(xrow + K_MAIN);
        t0 = *(const v2f*)(w0 + K_MAIN);
        t1 = *(const v2f*)(w1 + K_MAIN);
        t2 = *(const v2f*)(w2 + K_MAIN);
        t3 = *(const v2f*)(w3 + K_MAIN);
      }
      sumsq = fmaf(at.x, at.x, sumsq);
      sumsq = fmaf(at.y, at.y, sumsq);
      acc0 = wmma_f32_k4(at, t0, acc0);
      acc1 = wmma_f32_k4(at, t1, acc1);
      acc2 = wmma_f32_k4(at, t2, acc2);
      acc3 = wmma_f32_k4(at, t3, acc3);
    }

    float tot = sumsq + __shfl_xor(sumsq, 16, 32);
    if (lane < 16) s_sxp[wave][lane] = tot;

    // stash this wave's partial tile
    #pragma unroll
    for (int j = 0; j < 8; ++j) {
      const int r = j + half * 8;        // C/D layout: M = j + 8*(lane>=16)
      s_part[wave][r][c0] = acc0[j];
      s_part[wave][r][c1] = acc1[j];
      s_part[wave][r][c2] = acc2[j];
      s_part[wave][r][c3] = acc3[j];
    }
    __syncthreads();
    // cross-wave reduction into s_mx for rows of this weight class
    {
      const int r  = t >> 3;
      const int cb = (t & 7) * 8;
      const int gr = row_base + r;
      const bool mine = (wsel == 0) ? (gr < N_DRUG) : (gr >= N_DRUG);
      if (mine) {
        #pragma unroll
        for (int i = 0; i < 8; ++i) {
          const int c = cb + i;
          s_mx[r][c] = (s_part[0][r][c] + s_part[1][r][c]) +
                       (s_part[2][r][c] + s_part[3][r][c]);
        }
      }
    }
    __syncthreads();                      // s_part reused by second pass
  }

  // ---- per-row reductions: ||mx||^2 and dot(mx, hb) ----
  const int r = t >> 3, seg = t & 7;
  {
    const int cls = ((row_base + r) < N_DRUG) ? 0 : 1;
    float ps = 0.f, pd = 0.f;
    #pragma unroll
    for (int i = 0; i < 8; ++i) {
      float m = s_mx[r][seg * 8 + i];
      ps = fmaf(m, m, ps);
      pd = fmaf(m, s_hb[cls][seg * 8 + i], pd);
    }
    s_ps[r][seg] = ps; s_pd[r][seg] = pd;
  }
  __syncthreads();

  // ---- per-row scalar chain: matvec+proj+mobius_add+proj+logmap0 ----
  if (t < 16) {
    const int gr  = row_base + t;
    const int cls = (gr < N_DRUG) ? 0 : 1;
    float mxn2 = 0.f, dotP = 0.f;
    #pragma unroll
    for (int i = 0; i < 8; ++i) { mxn2 += s_ps[t][i]; dotP += s_pd[t][i]; }
    float xn2 = (s_sxp[0][t] + s_sxp[1][t]) + (s_sxp[2][t] + s_sxp[3][t]);
    float xn  = fmaxf(sqrtf(xn2), MIN_NRM);
    float mxn = fmaxf(sqrtf(mxn2), MIN_NRM);
    float s1  = tanhf((mxn / xn) * artanh_f(xn)) / mxn;   // mv = s1 * mx
    float na_mx = sqrtf(mxn2);
    float rn = fabsf(s1) * na_mx;                         // ||mv||
    if (rn > MAXNORM) { s1 *= MAXNORM / rn; rn = MAXNORM; }
    float x2 = rn * rn;
    float y2 = s_hby2[cls];
    float xy = s1 * dotP;                                 // dot(mv, hb)
    float ca = 1.f + 2.f * xy + y2;
    float cb = 1.f - x2;
    float den = fmaxf(1.f + 2.f * xy + x2 * y2, MIN_NRM);
    float num2 = ca * ca * x2 + 2.f * ca * cb * xy + cb * cb * y2;
    float rn2 = sqrtf(fmaxf(num2, 0.f)) / den;            // ||mobius_add||
    float p2  = (rn2 > MAXNORM) ? (MAXNORM / rn2) : 1.f;  // proj
    float A1 = p2 * ca * s1 / den;
    float A2 = p2 * cb / den;
    float pn = fmaxf(p2 * rn2, MIN_NRM);                  // ||liner_h||
    float tl = artanh_f(pn) / pn;                         // logmap0 scale
    s_A1[t] = A1; s_A2[t] = A2; s_T1[t] = tl * A1; s_T2[t] = tl * A2;
  }
  __syncthreads();

  {
    const int gr = row_base + r;
    if (gr < N_NODES) {
      const int cls = (gr < N_DRUG) ? 0 : 1;
      const float A1 = s_A1[r], A2 = s_A2[r], T1 = s_T1[r], T2 = s_T2[r];
      #pragma unroll
      for (int i = 0; i < 8; ++i) {
        const int c = seg * 8 + i;
        float m = s_mx[r][c], hbv = s_hb[cls][c];
        liner_h[(size_t)gr * DIM_OUT + c] = fmaf(A1, m, A2 * hbv);
        x_tan  [(size_t)gr * DIM_OUT + c] = fmaf(T1, m, T2 * hbv);
      }
    }
  }
}

// ---------------------------------------------------------------------------
// Kernel 2: HypAgg. support_t = adj @ x_tangent, then per-row
// proj(expmap0(.)) collapses to a scalar scale.
// ---------------------------------------------------------------------------
__global__ __launch_bounds__(128)
void hgc_hyp_agg(const float* __restrict__ adj,
                 const float* __restrict__ x_tan,
                 float* __restrict__ agg_h) {
  __shared__ float s_part[4][16][65];
  __shared__ float s_st[16][65];
  __shared__ float s_ps[16][8];
  __shared__ float s_S[16];

  const int t    = threadIdx.x;
  const int lane = t & 31;
  const int wave = t >> 5;
  const int half = lane >> 4;
  const int l16  = lane & 15;
  const int row_base = blockIdx.x * 16;

  const int row_u = row_base + l16;
  const int row_a = (row_u < N_NODES) ? row_u : (N_NODES - 1);
  const float* arow = adj + (size_t)row_a * N_NODES;
  const int c0 = l16, c1 = 16 + l16, c2 = 32 + l16, c3 = 48 + l16;

  v8f acc0 = {0.f,0.f,0.f,0.f,0.f,0.f,0.f,0.f};
  v8f acc1 = acc0, acc2 = acc0, acc3 = acc0;

  // ---- software-pipelined main loop ----
  int k = wave * 4;
  v2f a, b0, b1, b2, b3;
  {
    const int kk = k + half * 2;
    a = *(const v2f*)(arow + kk);
    const float* p = x_tan + (size_t)kk * DIM_OUT;
    b0.x = p[c0];           b1.x = p[c1];
    b2.x = p[c2];           b3.x = p[c3];
    b0.y = p[DIM_OUT + c0]; b1.y = p[DIM_OUT + c1];
    b2.y = p[DIM_OUT + c2]; b3.y = p[DIM_OUT + c3];
  }
  for (; k + 16 < K_MAIN; k += 16) {
    const int kn = k + 16 + half * 2;     // prefetch next fragment
    const v2f a_n = *(const v2f*)(arow + kn);
    const float* p = x_tan + (size_t)kn * DIM_OUT;
    v2f b0_n, b1_n, b2_n, b3_n;
    b0_n.x = p[c0];           b1_n.x = p[c1];
    b2_n.x = p[c2];           b3_n.x = p[c3];
    b0_n.y = p[DIM_OUT + c0]; b1_n.y = p[DIM_OUT + c1];
    b2_n.y = p[DIM_OUT + c2]; b3_n.y = p[DIM_OUT + c3];
    acc0 = wmma_f32_k4(a, b0, acc0);
    acc1 = wmma_f32_k4(a, b1, acc1);
    acc2 = wmma_f32_k4(a, b2, acc2);
    acc3 = wmma_f32_k4(a, b3, acc3);
    a = a_n; b0 = b0_n; b1 = b1_n; b2 = b2_n; b3 = b3_n;
  }
  acc0 = wmma_f32_k4(a, b0, acc0);        // epilogue
  acc1 = wmma_f32_k4(a, b1, acc1);
  acc2 = wmma_f32_k4(a, b2, acc2);
  acc3 = wmma_f32_k4(a, b3, acc3);

  if (wave == 0) {                        // K tail (rows 1544,1545 valid)
    v2f at = {0.f,0.f}, t0 = {0.f,0.f}, t1 = {0.f,0.f},
        t2 = {0.f,0.f}, t3 = {0.f,0.f};
    if (half == 0) {
      at = *(const v2f*)(arow + K_MAIN);
      const float* p = x_tan + (size_t)K_MAIN * DIM_OUT;
      t0.x = p[c0];           t1.x = p[c1];
      t2.x = p[c2];           t3.x = p[c3];
      t0.y = p[DIM_OUT + c0]; t1.y = p[DIM_OUT + c1];
      t2.y = p[DIM_OUT + c2]; t3.y = p[DIM_OUT + c3];
    }
    acc0 = wmma_f32_k4(at, t0, acc0);
    acc1 = wmma_f32_k4(at, t1, acc1);
    acc2 = wmma_f32_k4(at, t2, acc2);
    acc3 = wmma_f32_k4(at, t3, acc3);
  }

  #pragma unroll
  for (int j = 0; j < 8; ++j) {
    const int r = j + half * 8;
    s_part[wave][r][c0] = acc0[j];
    s_part[wave][r][c1] = acc1[j];
    s_part[wave][r][c2] = acc2[j];
    s_part[wave][r][c3] = acc3[j];
  }
  __syncthreads();

  const int r = t >> 3, seg = t & 7;
  {
    float ps = 0.f;
    #pragma unroll
    for (int i = 0; i < 8; ++i) {
      const int c = seg * 8 + i;
      float v = (s_part[0][r][c] + s_part[1][r][c]) +
                (s_part[2][r][c] + s_part[3][r][c]);
      s_st[r][c] = v;
      ps = fmaf(v, v, ps);
    }
    s_ps[r][seg] = ps;
  }
  __syncthreads();
  if (t < 16) {
    float s2 = 0.f;
    #pragma unroll
    for (int i = 0; i < 8; ++i) s2 += s_ps[t][i];
    float na = sqrtf(s2);
    float un = fmaxf(na, MIN_NRM);
    float e  = tanhf(un) / un;            // expmap0
    float nh = e * na;
    if (nh > MAXNORM) e *= MAXNORM / nh;  // proj
    s_S[t] = e;
  }
  __syncthreads();
  {
    const int gr = row_base + r;
    if (gr < N_NODES) {
      const float S = s_S[r];
      #pragma unroll
      for (int i = 0; i < 8; ++i) {
        const int c = seg * 8 + i;
        agg_h[(size_t)gr * DIM_OUT + c] = S * s_st[r][c];
      }
    }
  }
}

// ---------------------------------------------------------------------------
// Kernel 3: HypAct. delt1 = relu([agg|liner] @ weightnode + biasnode);
// h = delt1*agg + (1-delt1)*liner.  K=128, split at the concat boundary so
// both halves are branch-free and fully unrollable.
// ---------------------------------------------------------------------------
__global__ __launch_bounds__(128)
void hgc_hyp_act(const float* __restrict__ agg_h,
                 const float* __restrict__ liner_h,
                 const float* __restrict__ wn,       // [128,64] row-major
                 const float* __restrict__ biasnode, // [1546,1]
                 float* __restrict__ out) {
  const int t    = threadIdx.x;
  const int lane = t & 31;
  const int wave = t >> 5;
  const int half = lane >> 4;
  const int l16  = lane & 15;
  const int row_base = blockIdx.x * 16;
  const int n_off    = wave * 16;

  const int row_u = row_base + l16;
  const int row_a = (row_u < N_NODES) ? row_u : (N_NODES - 1);
  const float* ra = agg_h   + (size_t)row_a * DIM_OUT;
  const float* rl = liner_h + (size_t)row_a * DIM_OUT;
  const int ncol  = n_off + l16;

  v8f acc = {0.f,0.f,0.f,0.f,0.f,0.f,0.f,0.f};
  #pragma unroll
  for (int k = 0; k < 64; k += 4) {       // zf cols 0..63 = agg_h
    const int kk = k + half * 2;
    const v2f a = *(const v2f*)(ra + kk);
    v2f b;
    b.x = wn[(size_t)kk * DIM_OUT + ncol];
    b.y = wn[(size_t)(kk + 1) * DIM_OUT + ncol];
    acc = wmma_f32_k4(a, b, acc);
  }
  #pragma unroll
  for (int k = 64; k < 128; k += 4) {     // zf cols 64..127 = liner_h
    const int kk = k + half * 2;
    const v2f a = *(const v2f*)(rl + (kk - 64));
    v2f b;
    b.x = wn[(size_t)kk * DIM_OUT + ncol];
    b.y = wn[(size_t)(kk + 1) * DIM_OUT + ncol];
    acc = wmma_f32_k4(a, b, acc);
  }

  #pragma unroll
  for (int j = 0; j < 8; ++j) {
    const int gr = row_base + j + half * 8;
    if (gr < N_NODES) {
      const float ag = agg_h  [(size_t)gr * DIM_OUT + ncol];
      const float lh = liner_h[(size_t)gr * DIM_OUT + ncol];
      const float d  = fmaxf(acc[j] + biasnode[gr], 0.f);
      out[(size_t)gr * DIM_OUT + ncol] = fmaf(d, ag - lh, lh);
    }
  }
}

// ---------------------------------------------------------------------------
extern "C" void kernel_launch(void* const* d_in, const int* in_sizes, int n_in,
                              void* d_out, int out_size, void* d_ws,
                              size_t ws_size, hipStream_t stream) {
  const float* x    = (const float*)d_in[0];
  const float* adj  = (const float*)d_in[1];
  const float* Wd   = (const float*)d_in[2];
  const float* Wm   = (const float*)d_in[3];
  const float* bd   = (const float*)d_in[4];
  const float* bm   = (const float*)d_in[5];
  const float* wn   = (const float*)d_in[6];
  const float* bnod = (const float*)d_in[7];
  float* out = (float*)d_out;
  float* ws  = (float*)d_ws;

  const size_t HSZ = (size_t)N_NODES * DIM_OUT;   // 98944 floats
  float* liner = ws;
  float* xtan  = ws + HSZ;
  float* aggh  = ws + 2 * HSZ;

  hgc_hyp_linear<<<M_TILES, 128, 0, stream>>>(x, Wd, Wm, bd, bm, liner, xtan);
  hgc_hyp_agg   <<<M_TILES, 128, 0, stream>>>(adj, xtan, aggh);
  hgc_hyp_act   <<<M_TILES, 128, 0, stream>>>(aggh, liner, wn, bnod, out);
  // second tuple element: adj passthrough
  hipMemcpyAsync(out + HSZ, adj, (size_t)N_NODES * N_NODES * sizeof(float),
                 hipMemcpyDeviceToDevice, stream);
}